// Mamba2Simple_37546604101934
// MI455X (gfx1250) — compile-verified
//
#include <hip/hip_runtime.h>
#include <hip/hip_bf16.h>
#include <math.h>
#include <stdint.h>

typedef __attribute__((ext_vector_type(2))) float v2f;
typedef __attribute__((ext_vector_type(4))) float v4f;
typedef __attribute__((ext_vector_type(8))) float v8f;
typedef __attribute__((ext_vector_type(4))) int   v4i;

#define D_MODEL   384
#define D_STATE   64
#define D_CONV    4
#define HEADDIM   64
#define D_INNER   768
#define NHEADS    12
#define D_IN_PROJ 1676
#define D_CONV_IN 896
#define SEQLEN    4096
#define BATCH     8
#define NTOK      (BATCH * SEQLEN)
#define EPS       1e-5f

// GEMM tiling
#define BM   64
#define BN   128
#define KC   32
#define LDSS 36   // padded row stride (floats): 144B rows -> 16B aligned, bank-spread

#if __has_builtin(__builtin_amdgcn_global_load_async_to_lds_b128) && \
    __has_builtin(__builtin_amdgcn_s_wait_asynccnt)
#define USE_ASYNC_LDS 1
#else
#define USE_ASYNC_LDS 0
#endif

// Typed address-space pointers for the async builtin:
//   arg0: addrspace(1) v4i*  (global source)
//   arg1: addrspace(3) v4i*  (LDS destination; low 32 bits of generic addr)
typedef __attribute__((address_space(1))) v4i gas_v4i;
typedef __attribute__((address_space(3))) v4i las_v4i;
#define AS1(p) ((gas_v4i*)(uintptr_t)(p))
#define AS3(p) ((las_v4i*)(uint32_t)(uintptr_t)(p))

// ---------------------------------------------------------------------------
// FP32 GEMM via V_WMMA_F32_16X16X4_F32, LDS-staged & double-buffered.
// D[M,N] = A[M,K] * W[N,K]^T  (row-major A and W).
// Block: 256 thr = 8 waves (4 M-waves x 2 N-waves), block tile 64x128xKc=32.
// Producer path: GLOBAL_LOAD_ASYNC_TO_LDS_B128 (ASYNCcnt) when available, so
// the next chunk streams into LDS while 32 WMMAs run on the current one.
// ---------------------------------------------------------------------------
__global__ __launch_bounds__(256) void gemm_wmma_f32(
    const float* __restrict__ A, const float* __restrict__ W,
    float* __restrict__ D, int M, int N, int K, int lda, int ldw, int ldd)
{
  __shared__ float As[2][BM][LDSS];
  __shared__ float Bs[2][BN][LDSS];

  const int tid   = threadIdx.x;
  const int lane  = tid & 31;
  const int wave  = tid >> 5;
  const int waveM = wave >> 1;  // 0..3
  const int waveN = wave & 1;   // 0..1
  const int mBlock = blockIdx.y * BM;
  const int nBlock = blockIdx.x * BN;

  // Tile-slot mapping: v4f slots; A tile 64x32 = 512 slots (2/thr),
  // B tile 128x32 = 1024 slots (4/thr). slot s -> row s/8, col (s%8)*4.
  const int gRow = tid >> 3;        // 0..31
  const int gCol = (tid & 7) * 4;   // 0,4,..,28

#if USE_ASYNC_LDS
  auto issueAsync = [&](int buf, int k0) {
#pragma unroll
    for (int i = 0; i < 2; ++i) {
      const int row = gRow + i * 32;
      const float* g = A + (size_t)(mBlock + row) * lda + k0 + gCol;
      __builtin_amdgcn_global_load_async_to_lds_b128(AS1(g), AS3(&As[buf][row][gCol]), 0, 0);
    }
#pragma unroll
    for (int i = 0; i < 4; ++i) {
      const int row  = gRow + i * 32;                 // 0..127 = tile column
      const int col  = nBlock + row;
      const int crow = col < N ? col : N - 1;         // clamp, never stored
      const float* g = W + (size_t)crow * ldw + k0 + gCol;
      __builtin_amdgcn_global_load_async_to_lds_b128(AS1(g), AS3(&Bs[buf][row][gCol]), 0, 0);
    }
  };
#else
  v4f aR[2], bR[4];
  auto loadGlobal = [&](int k0) {
#pragma unroll
    for (int i = 0; i < 2; ++i) {
      const int row = gRow + i * 32;
      aR[i] = *(const v4f*)(A + (size_t)(mBlock + row) * lda + k0 + gCol);
    }
#pragma unroll
    for (int i = 0; i < 4; ++i) {
      const int row  = gRow + i * 32;
      const int col  = nBlock + row;
      const int crow = col < N ? col : N - 1;
      bR[i] = *(const v4f*)(W + (size_t)crow * ldw + k0 + gCol);
    }
  };
  auto storeLds = [&](int buf) {
#pragma unroll
    for (int i = 0; i < 2; ++i)
      *(v4f*)&As[buf][gRow + i * 32][gCol] = aR[i];
#pragma unroll
    for (int i = 0; i < 4; ++i)
      *(v4f*)&Bs[buf][gRow + i * 32][gCol] = bR[i];
  };
#endif

  // WMMA fragment lane mapping (16x4 f32 A / 4x16 f32 B):
  // lanes 0-15: M/N = lane, K = k,k+1 ; lanes 16-31: M/N = lane-16, K = k+2,k+3
  const int laneM = lane & 15;
  const int kOff  = (lane >> 4) << 1;

  v8f acc0 = {}, acc1 = {}, acc2 = {}, acc3 = {};
  const int nk = K / KC;

  auto computeChunk = [&](int buf) {
    const float* aS  = &As[buf][waveM * 16 + laneM][kOff];
    const float* b0S = &Bs[buf][waveN * 64 +  0 + laneM][kOff];
    const float* b1S = &Bs[buf][waveN * 64 + 16 + laneM][kOff];
    const float* b2S = &Bs[buf][waveN * 64 + 32 + laneM][kOff];
    const float* b3S = &Bs[buf][waveN * 64 + 48 + laneM][kOff];
#pragma unroll
    for (int k = 0; k < KC; k += 4) {
      v2f a  = *(const v2f*)(aS  + k);
      v2f b0 = *(const v2f*)(b0S + k);
      v2f b1 = *(const v2f*)(b1S + k);
      v2f b2 = *(const v2f*)(b2S + k);
      v2f b3 = *(const v2f*)(b3S + k);
      acc0 = __builtin_amdgcn_wmma_f32_16x16x4_f32(false, a, false, b0, (short)0, acc0, false, false);
      acc1 = __builtin_amdgcn_wmma_f32_16x16x4_f32(false, a, false, b1, (short)0, acc1, false, false);
      acc2 = __builtin_amdgcn_wmma_f32_16x16x4_f32(false, a, false, b2, (short)0, acc2, false, false);
      acc3 = __builtin_amdgcn_wmma_f32_16x16x4_f32(false, a, false, b3, (short)0, acc3, false, false);
    }
  };

#if USE_ASYNC_LDS
  issueAsync(0, 0);
  __builtin_amdgcn_s_wait_asynccnt(0);
  __syncthreads();
  for (int kc = 0; kc < nk; ++kc) {
    const int buf = kc & 1;
    if (kc + 1 < nk) issueAsync(buf ^ 1, (kc + 1) * KC);  // stream next chunk
    computeChunk(buf);                                    // 32 WMMAs hide it
    __builtin_amdgcn_s_wait_asynccnt(0);                  // own writes done
    __syncthreads();                                      // everyone's done
  }
#else
  loadGlobal(0);
  storeLds(0);
  __syncthreads();
  for (int kc = 0; kc < nk; ++kc) {
    const int buf = kc & 1;
    if (kc + 1 < nk) loadGlobal((kc + 1) * KC);
    computeChunk(buf);
    __syncthreads();
    if (kc + 1 < nk) storeLds(buf ^ 1);
    __syncthreads();
  }
#endif

  // Store D: VGPR r, lanes 0-15 -> (M=r, N=lane); lanes 16-31 -> (M=r+8, N=lane-16)
  const int rOff = (lane >> 4) * 8;
  const int c0 = nBlock + waveN * 64 +  0 + laneM;
  const int c1 = nBlock + waveN * 64 + 16 + laneM;
  const int c2 = nBlock + waveN * 64 + 32 + laneM;
  const int c3 = nBlock + waveN * 64 + 48 + laneM;
#pragma unroll
  for (int r = 0; r < 8; ++r) {
    float* drow = D + (size_t)(mBlock + waveM * 16 + rOff + r) * ldd;
    if (c0 < N) drow[c0] = acc0[r];
    if (c1 < N) drow[c1] = acc1[r];
    if (c2 < N) drow[c2] = acc2[r];
    if (c3 < N) drow[c3] = acc3[r];
  }
}

// ---------------------------------------------------------------------------
// Fused: depthwise causal conv(4) + bias + SiLU over the 896 xBC channels,
// RMSNorm over the first 768 (x) channels, B/C extraction, dt softplus.
// One block per token, 128 threads x 7 strided channels.
// ---------------------------------------------------------------------------
__global__ __launch_bounds__(128) void conv_norm_kernel(
    const float* __restrict__ zxbcdt,
    const float* __restrict__ conv_w, const float* __restrict__ conv_b,
    const float* __restrict__ norm_w, const float* __restrict__ dt_bias,
    float* __restrict__ xn, float* __restrict__ Bm, float* __restrict__ Cm,
    float* __restrict__ dts)
{
  __shared__ float red[128];
  const int tok = blockIdx.x;
  const int l   = tok & (SEQLEN - 1);
  const int t   = threadIdx.x;
  const float* base = zxbcdt + (size_t)tok * D_IN_PROJ + D_INNER;  // xBC cols

  float v[7];
  float ss = 0.f;
#pragma unroll
  for (int kk = 0; kk < 7; ++kk) {
    const int c = t + kk * 128;
    float s = conv_b[c];
#pragma unroll
    for (int j = 0; j < D_CONV; ++j) {
      const int ll = l + j - (D_CONV - 1);
      if (ll >= 0)
        s += conv_w[c * D_CONV + j] * base[(ptrdiff_t)(j - (D_CONV - 1)) * D_IN_PROJ + c];
    }
    const float sv = s / (1.f + __expf(-s));  // SiLU
    v[kk] = sv;
    if (kk < 6) ss += sv * sv;                // x region: channels < 768
  }

  red[t] = ss;
  __syncthreads();
#pragma unroll
  for (int off = 64; off > 0; off >>= 1) {
    if (t < off) red[t] += red[t + off];
    __syncthreads();
  }
  const float rstd = rsqrtf(red[0] * (1.f / (float)D_INNER) + EPS);

#pragma unroll
  for (int kk = 0; kk < 6; ++kk) {
    const int c = t + kk * 128;
    xn[(size_t)tok * D_INNER + c] = v[kk] * rstd * norm_w[c];
  }
  // kk == 6: channels 768..895 -> B (first 64) then C (next 64)
  if (t < 64) Bm[(size_t)tok * 64 + t]        = v[6];
  else        Cm[(size_t)tok * 64 + (t - 64)] = v[6];

  if (t < NHEADS) {
    const float raw = zxbcdt[(size_t)tok * D_IN_PROJ + (D_IN_PROJ - NHEADS) + t] + dt_bias[t];
    dts[(size_t)tok * NHEADS + t] = (raw > 20.f) ? raw : log1pf(__expf(raw));
  }
}

// ---------------------------------------------------------------------------
// SSD sequential scan. One block per (batch, head); 64 threads, lane = p.
// 64-entry state column per lane lives in VGPRs; B/C broadcast via LDS.
// Fuses y = (C.h + x*Dp) * silu(z) and writes the out_proj input.
// ---------------------------------------------------------------------------
__global__ __launch_bounds__(64) void ssd_scan_kernel(
    const float* __restrict__ xn, const float* __restrict__ Bm,
    const float* __restrict__ Cm, const float* __restrict__ dts,
    const float* __restrict__ zxbcdt,
    const float* __restrict__ A_log, const float* __restrict__ Dp,
    float* __restrict__ yz)
{
  __shared__ float sB[64];
  __shared__ float sC[64];
  const int bh = blockIdx.x;
  const int b  = bh / NHEADS;
  const int h  = bh % NHEADS;
  const int p  = threadIdx.x;

  const float Ah = -__expf(A_log[h]);
  const float Dh = Dp[h];

  float hs[D_STATE];
#pragma unroll
  for (int n = 0; n < D_STATE; ++n) hs[n] = 0.f;

  size_t tok = (size_t)b * SEQLEN;
  for (int l = 0; l < SEQLEN; ++l, ++tok) {
    const float dtv = dts[tok * NHEADS + h];
    const float dA  = __expf(dtv * Ah);
    const float x   = xn[tok * D_INNER + h * HEADDIM + p];
    const float bb  = Bm[tok * 64 + p] * dtv;
    const float cc  = Cm[tok * 64 + p];

    __syncthreads();
    sB[p] = bb;
    sC[p] = cc;
    __syncthreads();

    float acc = 0.f;
#pragma unroll 8
    for (int n = 0; n < D_STATE; ++n) {
      const float hn = hs[n] * dA + sB[n] * x;
      hs[n] = hn;
      acc += sC[n] * hn;
    }

    const float y  = acc + x * Dh;
    const float zv = zxbcdt[tok * D_IN_PROJ + h * HEADDIM + p];
    yz[tok * D_INNER + h * HEADDIM + p] = y * (zv / (1.f + __expf(-zv)));
  }
}

// ---------------------------------------------------------------------------
extern "C" void kernel_launch(void* const* d_in, const int* in_sizes, int n_in,
                              void* d_out, int out_size, void* d_ws, size_t ws_size,
                              hipStream_t stream) {
  const float* u          = (const float*)d_in[0];
  const float* in_proj_w  = (const float*)d_in[1];
  const float* conv_w     = (const float*)d_in[2];
  const float* conv_b     = (const float*)d_in[3];
  const float* norm_w     = (const float*)d_in[4];
  const float* out_proj_w = (const float*)d_in[5];
  const float* A_log      = (const float*)d_in[6];
  const float* Dpv        = (const float*)d_in[7];
  const float* dt_bias    = (const float*)d_in[8];
  float* out = (float*)d_out;

  float* zxbcdt = (float*)d_ws;                              // NTOK * 1676
  float* xn     = zxbcdt + (size_t)NTOK * D_IN_PROJ;         // NTOK * 768
  float* Bm     = xn     + (size_t)NTOK * D_INNER;           // NTOK * 64
  float* Cm     = Bm     + (size_t)NTOK * 64;                // NTOK * 64
  float* dts    = Cm     + (size_t)NTOK * 64;                // NTOK * 12
  float* yz     = dts    + (size_t)NTOK * NHEADS;            // NTOK * 768

  // 1) zxbcdt = u @ in_proj_w^T
  dim3 g1((D_IN_PROJ + BN - 1) / BN, NTOK / BM);
  gemm_wmma_f32<<<g1, 256, 0, stream>>>(u, in_proj_w, zxbcdt,
                                        NTOK, D_IN_PROJ, D_MODEL,
                                        D_MODEL, D_MODEL, D_IN_PROJ);

  // 2) conv + SiLU + RMSNorm + splits + softplus(dt)
  conv_norm_kernel<<<NTOK, 128, 0, stream>>>(zxbcdt, conv_w, conv_b, norm_w,
                                             dt_bias, xn, Bm, Cm, dts);

  // 3) sequential SSD scan, fused with gating by silu(z)
  ssd_scan_kernel<<<BATCH * NHEADS, 64, 0, stream>>>(xn, Bm, Cm, dts, zxbcdt,
                                                     A_log, Dpv, yz);

  // 4) out = yz @ out_proj_w^T
  dim3 g2((D_MODEL + BN - 1) / BN, NTOK / BM);
  gemm_wmma_f32<<<g2, 256, 0, stream>>>(yz, out_proj_w, out,
                                        NTOK, D_MODEL, D_INNER,
                                        D_INNER, D_INNER, D_MODEL);
}